// FrameAlignment_39771397161407
// MI455X (gfx1250) — compile-verified
//
#include <hip/hip_runtime.h>
#include <math.h>

typedef __attribute__((ext_vector_type(2))) float v2f;
typedef __attribute__((ext_vector_type(4))) float v4f;
typedef __attribute__((ext_vector_type(8))) float v8f;

#define Bn 8
#define Sn 2048
#define Fn 128
#define Wn 5
#define NW (Sn - Wn + 1)   // 2044
#define KC 64

// ------------------------------------------------------------------------
// Fused online-softmax + GEMM (flash style): ONE pass over proj.
// Block = 256 threads (8 waves); block owns 16 rows x 128 cols of `aligned`.
// Producer mapping: thread t -> tile row r = t>>4, cols c = (t&15)*4.
// The 16 producer threads of row r occupy one lane-half of wave r>>1, so
// per-row reductions are intra-half __shfl_xor and the running softmax
// state (m, s) is kept redundantly in registers by all 16 threads.
// Consumer mapping: wave w computes 16x16 WMMA tile at cols 16w, rescaling
// its accumulator by exp(m_old - m_new) per K-chunk.
// ------------------------------------------------------------------------
__global__ void __launch_bounds__(256)
k_flash_softmax_gemm(const float* __restrict__ proj, const float* __restrict__ score,
                     const float* __restrict__ feature, float* __restrict__ out) {
    __shared__ float ldsA[16][KC + 1];          // softmax weights, +1 pad
    __shared__ __align__(16) float smScale[16]; // per-row acc rescale factor
    __shared__ __align__(16) float smInv[16];   // per-row 1/sum at the end

    const int bid  = blockIdx.x;
    const int b    = bid >> 7;            // / (Sn/16)
    const int i0   = (bid & 127) << 4;
    const int tid  = threadIdx.x;
    const int lane = tid & 31;
    const int wave = tid >> 5;
    const int n0   = wave << 4;           // this wave's 16 output columns

    // producer mapping
    const int r = tid >> 4;               // tile row 0..15
    const int c = (tid & 15) << 2;        // col group, float4
    const int rowIdx = b * Sn + i0 + r;
    const float* prow  = proj    + (size_t)rowIdx * Sn;
    const float* srow  = score   + (size_t)b * Sn;
    const float* fbase = feature + (size_t)b * Sn * Fn;

    // consumer (WMMA) lane mapping
    const int m16   = lane & 15;
    const int half  = lane >> 4;          // 0 or 1
    const int khalf = half << 1;          // 0 or 2

    float m = -INFINITY, s = 0.0f;        // running softmax state (per row)
    v8f acc = {};

    for (int k0 = 0; k0 < Sn; k0 += KC) {
        // ---- producer: online softmax of this 16 x KC chunk -------------
        v4f p  = *(const v4f*)(prow + k0 + c);
        v4f sc = *(const v4f*)(srow + k0 + c);
        float x0 = p[0] * sc[0], x1 = p[1] * sc[1];
        float x2 = p[2] * sc[2], x3 = p[3] * sc[3];
        float cm = fmaxf(fmaxf(x0, x1), fmaxf(x2, x3));
        #pragma unroll
        for (int off = 1; off < 16; off <<= 1)
            cm = fmaxf(cm, __shfl_xor(cm, off, 32));   // stays in lane-half

        const float mnew  = fmaxf(m, cm);
        const float scale = __expf(m - mnew);          // first chunk: exp(-inf)=0
        float w0 = __expf(x0 - mnew), w1 = __expf(x1 - mnew);
        float w2 = __expf(x2 - mnew), w3 = __expf(x3 - mnew);
        float csum = (w0 + w1) + (w2 + w3);
        #pragma unroll
        for (int off = 1; off < 16; off <<= 1)
            csum += __shfl_xor(csum, off, 32);
        s = s * scale + csum;
        m = mnew;

        ldsA[r][c + 0] = w0;
        ldsA[r][c + 1] = w1;
        ldsA[r][c + 2] = w2;
        ldsA[r][c + 3] = w3;
        if ((tid & 15) == 0) smScale[r] = scale;
        __syncthreads();

        // ---- consumer: rescale accumulator by this chunk's row scales ---
        v4f s0 = *(const v4f*)&smScale[half * 8];
        v4f s1 = *(const v4f*)&smScale[half * 8 + 4];
        #pragma unroll
        for (int rr = 0; rr < 4; ++rr) { acc[rr] *= s0[rr]; acc[rr + 4] *= s1[rr]; }

        // ---- consumer: 16 fp32 WMMAs over the KC=64 chunk ----------------
        #pragma unroll
        for (int t = 0; t < KC / 4; ++t) {
            // A (16x4 f32): v0 = K {0|2}, v1 = K {1|3} per lane-half
            v2f a;
            a.x = ldsA[m16][4 * t + khalf];
            a.y = ldsA[m16][4 * t + khalf + 1];
            // B (4x16 f32): row K striped across 16 lanes per half
            const float* fp = fbase + (size_t)(k0 + 4 * t + khalf) * Fn + n0 + m16;
            v2f bb;
            bb.x = fp[0];
            bb.y = fp[Fn];
            acc = __builtin_amdgcn_wmma_f32_16x16x4_f32(
                false, a, false, bb, (short)0, acc, false, false);
        }
        __syncthreads();
    }

    // ---- final normalization by 1/sum and store --------------------------
    if ((tid & 15) == 0) smInv[r] = 1.0f / s;
    __syncthreads();
    v4f i0v = *(const v4f*)&smInv[half * 8];
    v4f i1v = *(const v4f*)&smInv[half * 8 + 4];
    #pragma unroll
    for (int rr = 0; rr < 4; ++rr) { acc[rr] *= i0v[rr]; acc[rr + 4] *= i1v[rr]; }

    // C/D 16x16 f32: VGPR rr -> rows rr (lanes 0-15) and rr+8 (lanes 16-31)
    const int orow0 = i0 + (half << 3);
    float* obase = out + (size_t)b * Sn * Fn + (size_t)orow0 * Fn + n0 + m16;
    #pragma unroll
    for (int rr = 0; rr < 8; ++rr)
        obase[(size_t)rr * Fn] = acc[rr];
}

// -------------------------------------------- per-row dot / squared norms
__global__ void __launch_bounds__(256)
k_rowdots(const float* __restrict__ feature, const float* __restrict__ aligned,
          float* __restrict__ dotv, float* __restrict__ nfv, float* __restrict__ nav) {
    const int row  = blockIdx.x * 8 + (threadIdx.x >> 5);
    const int lane = threadIdx.x & 31;
    v4f fv = *(const v4f*)(feature + (size_t)row * Fn + lane * 4);
    v4f av = *(const v4f*)(aligned + (size_t)row * Fn + lane * 4);
    float d = 0.f, nf = 0.f, na = 0.f;
    #pragma unroll
    for (int q = 0; q < 4; ++q) {
        d  += fv[q] * av[q];
        nf += fv[q] * fv[q];
        na += av[q] * av[q];
    }
    #pragma unroll
    for (int off = 16; off > 0; off >>= 1) {
        d  += __shfl_xor(d,  off, 32);
        nf += __shfl_xor(nf, off, 32);
        na += __shfl_xor(na, off, 32);
    }
    if (lane == 0) { dotv[row] = d; nfv[row] = nf; nav[row] = na; }
}

// ------------------------------------------------- windowed cosine loss
__global__ void __launch_bounds__(256)
k_window(const float* __restrict__ dotv, const float* __restrict__ nfv,
         const float* __restrict__ nav, float* __restrict__ part) {
    __shared__ float red[256];
    const int w = blockIdx.x * 256 + threadIdx.x;
    float v = 0.0f;
    if (w < Bn * NW) {
        const int b = w / NW;
        const int i = w % NW;
        const int base = b * Sn + i;
        float dw = 0.f, fw = 0.f, aw = 0.f;
        #pragma unroll
        for (int j = 0; j < Wn; ++j) {
            dw += dotv[base + j];
            fw += nfv[base + j];
            aw += nav[base + j];
        }
        float cs = dw / (fmaxf(sqrtf(fw), 1e-12f) * fmaxf(sqrtf(aw), 1e-12f));
        v = 1.0f - cs;
    }
    red[threadIdx.x] = v;
    __syncthreads();
    #pragma unroll
    for (int off = 128; off > 0; off >>= 1) {
        if (threadIdx.x < off) red[threadIdx.x] += red[threadIdx.x + off];
        __syncthreads();
    }
    if (threadIdx.x == 0) part[blockIdx.x] = red[0];
}

__global__ void k_final(const float* __restrict__ part, float* __restrict__ loss_out) {
    float s = 0.0f;
    for (int i = 0; i < 64; ++i) s += part[i];   // fixed order -> deterministic
    loss_out[0] = s / (float)(Bn * NW);
}

// ---------------------------------------------------------------- launcher
extern "C" void kernel_launch(void* const* d_in, const int* in_sizes, int n_in,
                              void* d_out, int out_size, void* d_ws, size_t ws_size,
                              hipStream_t stream) {
    const float* score   = (const float*)d_in[0];   // [B,S,1]
    const float* feature = (const float*)d_in[1];   // [B,S,F]
    const float* proj    = (const float*)d_in[2];   // [B,S,S]
    float* out = (float*)d_out;                     // aligned [B,S,F] ++ loss [1]
    float* ws  = (float*)d_ws;

    float* dotv = ws;                   // B*S = 16384 each
    float* nfv  = ws + 16384;
    float* nav  = ws + 32768;
    float* part = ws + 49152;           // 64 block partials

    k_flash_softmax_gemm<<<Bn * (Sn / 16), 256, 0, stream>>>(proj, score, feature, out);
    k_rowdots           <<<(Bn * Sn) / 8, 256, 0, stream>>>(feature, out, dotv, nfv, nav);
    k_window            <<<64, 256, 0, stream>>>(dotv, nfv, nav, part);
    k_final             <<<1, 1, 0, stream>>>(part, out + (size_t)Bn * Sn * Fn);
}